// SingleRoIExtractor_11029476016431
// MI455X (gfx1250) — compile-verified
//
#include <hip/hip_runtime.h>
#include <hip/hip_bf16.h>

typedef __attribute__((ext_vector_type(2))) float v2f;
typedef __attribute__((ext_vector_type(8))) float v8f;

typedef const float __attribute__((address_space(1)))* gfp;
typedef const char  __attribute__((address_space(1)))* gcp;

#define ROI_OUT 7
#define NBIN    49   // 7*7
#define CCH     256

__device__ __forceinline__ float gload(gfp base, unsigned boff) {
    return *(gfp)((gcp)base + boff);
}

// One block (8 wave32) per roi.
// Phase 1: per-bin bilinear tap table (16 taps/bin = 4 samples x 4 corners):
//          weights pre-scaled by 0.25 (sample mean, zeroed when invalid) and
//          tap offsets stored as BYTE offsets within one channel plane.
// Phase 2: each wave handles channels {16w..16w+15} and {+128}; per bin,
//          out[c] = sum_{j<16} w_j * feat[c, off_j] via 4 accumulating
//          V_WMMA_F32_16X16X4_F32 (A = weight chunk broadcast to all rows,
//          B = gathered 4x16 tap block). Two independent acc chains share A.
__global__ __launch_bounds__(256) void roi_align_fpn_wmma_kernel(
    const float* __restrict__ f0, const float* __restrict__ f1,
    const float* __restrict__ f2, const float* __restrict__ f3,
    const float* __restrict__ rois, float* __restrict__ out, int K)
{
    __shared__ float    sw[NBIN * 16];
    __shared__ unsigned sob[NBIN * 16];

    const int k = blockIdx.x;
    if (k >= K) return;
    const int tid = threadIdx.x;

    // ---- per-roi scalar setup ----
    const float* roi = rois + 5 * k;
    const float bxf = roi[0];
    const float rx1 = roi[1], ry1 = roi[2], rx2 = roi[3], ry2 = roi[4];

    const float scale = sqrtf((rx2 - rx1 + 1.0f) * (ry2 - ry1 + 1.0f));
    int lvl = (int)floorf(log2f(scale * (1.0f / 56.0f) + 1e-6f));
    lvl = lvl < 0 ? 0 : (lvl > 3 ? 3 : lvl);

    // Force wave-uniform (SGPR) level / batch index -> scalar base address.
    const int lvl_u = __builtin_amdgcn_readfirstlane(lvl);
    const int bi_u  = __builtin_amdgcn_readfirstlane((int)bxf);

    const int   W  = (lvl_u == 0) ? 336 : (lvl_u == 1) ? 168 : (lvl_u == 2) ? 84 : 42;
    const int   H  = (lvl_u == 0) ? 200 : (lvl_u == 1) ? 100 : (lvl_u == 2) ? 50 : 25;
    const float sc = (lvl_u == 0) ? 0.25f : (lvl_u == 1) ? 0.125f
                                          : (lvl_u == 2) ? 0.0625f : 0.03125f;
    const float* fb = (lvl_u == 0) ? f0 : (lvl_u == 1) ? f1 : (lvl_u == 2) ? f2 : f3;
    const unsigned HW = (unsigned)(H * W);

    // Uniform base: start of this batch's channel planes (addrspace(1)).
    const gfp fg = (gfp)(unsigned long long)(fb + (size_t)bi_u * (size_t)CCH * HW);

    const float x1  = rx1 * sc, y1v = ry1 * sc, x2 = rx2 * sc, y2v = ry2 * sc;
    const float roiw = fmaxf(x2 - x1, 1.0f);
    const float roih = fmaxf(y2v - y1v, 1.0f);
    const float bw = roiw * (1.0f / ROI_OUT);
    const float bh = roih * (1.0f / ROI_OUT);

    // ---- phase 1: build tap table ----
    if (tid < NBIN * 4) {
        const int bin = tid >> 2;
        const int s   = tid & 3;
        const int ph = bin / ROI_OUT, pw = bin % ROI_OUT;
        const int iy = s >> 1, ix = s & 1;

        const float y = y1v + ((float)ph + ((float)iy + 0.5f) * 0.5f) * bh;
        const float x = x1  + ((float)pw + ((float)ix + 0.5f) * 0.5f) * bw;

        const bool valid = (y > -1.0f) && (y < (float)H) && (x > -1.0f) && (x < (float)W);
        const float yc = fminf(fmaxf(y, 0.0f), (float)(H - 1));
        const float xc = fminf(fmaxf(x, 0.0f), (float)(W - 1));
        const float yf = floorf(yc), xf = floorf(xc);
        const float ly = yc - yf, lx = xc - xf;
        const float hy = 1.0f - ly, hx = 1.0f - lx;
        const int y0i = (int)yf, x0i = (int)xf;
        const int y1i = min(y0i + 1, H - 1);
        const int x1i = min(x0i + 1, W - 1);
        const float m = valid ? 0.25f : 0.0f;

        const int base = bin * 16 + s * 4;
        sw[base + 0] = hy * hx * m;  sob[base + 0] = (unsigned)(y0i * W + x0i) << 2;
        sw[base + 1] = hy * lx * m;  sob[base + 1] = (unsigned)(y0i * W + x1i) << 2;
        sw[base + 2] = ly * hx * m;  sob[base + 2] = (unsigned)(y1i * W + x0i) << 2;
        sw[base + 3] = ly * lx * m;  sob[base + 3] = (unsigned)(y1i * W + x1i) << 2;
    }
    __syncthreads();

    // ---- phase 2: WMMA tap reduction, two channel groups per wave ----
    const int wave  = tid >> 5;
    const int lane  = tid & 31;
    const int nsub  = lane & 15;   // N column / channel-in-group
    const int khalf = lane >> 4;   // lanes 0-15: K rows {0,1}; 16-31: {2,3}

    const int ch0 = wave * 16 + nsub;       // channels 0..127 across 8 waves
    const int ch1 = ch0 + 128;              // channels 128..255
    const unsigned co0 = ((unsigned)ch0 * HW) << 2;   // byte offset of channel plane
    const unsigned co1 = ((unsigned)ch1 * HW) << 2;

    float* oc0 = out + ((size_t)k * CCH + ch0) * NBIN;
    float* oc1 = oc0 + (size_t)128 * NBIN;

    __builtin_prefetch((const void*)((const char*)(fb + (size_t)bi_u * (size_t)CCH * HW) + co0 + sob[0]), 0, 3);
    __builtin_prefetch((const void*)((const char*)(fb + (size_t)bi_u * (size_t)CCH * HW) + co1 + sob[0]), 0, 3);

    for (int bin = 0; bin < NBIN; ++bin) {
        v8f acc0 = {};
        v8f acc1 = {};
#pragma unroll
        for (int t = 0; t < 4; ++t) {
            const int j = bin * 16 + t * 4 + khalf * 2;
            v2f a;
            a.x = sw[j];
            a.y = sw[j + 1];
            const unsigned o0 = sob[j];
            const unsigned o1 = sob[j + 1];
            v2f b0, b1;
            b0.x = gload(fg, co0 + o0);
            b0.y = gload(fg, co0 + o1);
            b1.x = gload(fg, co1 + o0);
            b1.y = gload(fg, co1 + o1);
            acc0 = __builtin_amdgcn_wmma_f32_16x16x4_f32(
                false, a, false, b0, (short)0, acc0, false, false);
            acc1 = __builtin_amdgcn_wmma_f32_16x16x4_f32(
                false, a, false, b1, (short)0, acc1, false, false);
        }
        // Row 0 of D: VGPR0 of lanes 0..15 (lanes 16..31 hold duplicate row 8).
        if (lane < 16) {
            oc0[bin] = acc0[0];
            oc1[bin] = acc1[0];
        }
    }
}

extern "C" void kernel_launch(void* const* d_in, const int* in_sizes, int n_in,
                              void* d_out, int out_size, void* d_ws, size_t ws_size,
                              hipStream_t stream) {
    (void)n_in; (void)out_size; (void)d_ws; (void)ws_size;
    const float* f0   = (const float*)d_in[0];
    const float* f1   = (const float*)d_in[1];
    const float* f2   = (const float*)d_in[2];
    const float* f3   = (const float*)d_in[3];
    const float* rois = (const float*)d_in[4];
    float* out = (float*)d_out;
    const int K = in_sizes[4] / 5;

    roi_align_fpn_wmma_kernel<<<K, 256, 0, stream>>>(f0, f1, f2, f3, rois, out, K);
}